// VectorQuantGroup_1451698946507
// MI455X (gfx1250) — compile-verified
//
#include <hip/hip_runtime.h>

#define M_ROWS 32768   // 64*512*1
#define EMB    400
#define NCLS   2048
#define NGRP   8
#define CPG    256
#define NSAMP  10

typedef __attribute__((ext_vector_type(2))) float v2f;
typedef __attribute__((ext_vector_type(8))) float v8f;

// ---------------------------------------------------------------------------
// Kernel 0: zero histogram + SSE accumulator
// ---------------------------------------------------------------------------
__global__ void init_kernel(unsigned int* __restrict__ hist, double* __restrict__ sse) {
    int i = blockIdx.x * 256 + threadIdx.x;
    if (i < NCLS) hist[i] = 0u;
    if (i == 0)   *sse = 0.0;
}

// ---------------------------------------------------------------------------
// Kernel 1: per-row sum of squares (used for x rows and emb rows)
// ---------------------------------------------------------------------------
__global__ __launch_bounds__(128)
void rowsq_kernel(const float* __restrict__ a, float* __restrict__ out, int cols) {
    const int row = blockIdx.x;
    const int t   = threadIdx.x;
    const float* ar = a + (size_t)row * cols;
    float local = 0.0f;
    for (int c = t; c < cols; c += 128) { float v = ar[c]; local += v * v; }
    __shared__ float red[128];
    red[t] = local; __syncthreads();
    for (int s = 64; s > 0; s >>= 1) { if (t < s) red[t] += red[t + s]; __syncthreads(); }
    if (t == 0) out[row] = red[0];
}

// ---------------------------------------------------------------------------
// Kernel 2: FP32 WMMA GEMM  d = xsq + esq - 2 * x @ emb^T ;  enc = 1/d
// One wave computes a 32(M) x 64(N) tile: 8 accumulators, 2 A fragments
// reused across 4 B fragments (8 wmma per 6 loads -> load-bound no more).
// A 16x4 f32 layout: lane<16 -> {K0,K1}, lane>=16 -> {K2,K3}; row = lane&15.
// B 4x16 mirrors A with class = col. C: VGPR v -> M = v + 8*(lane>=16).
// ---------------------------------------------------------------------------
__global__ __launch_bounds__(256)
void gemm_dist_kernel(const float* __restrict__ x, const float* __restrict__ emb,
                      const float* __restrict__ xsq, const float* __restrict__ esq,
                      float* __restrict__ enc)
{
    const int lane    = threadIdx.x & 31;
    const int wave    = threadIdx.x >> 5;
    const int job     = blockIdx.x * 8 + wave;     // 32768 jobs total
    const int rowPair = job >> 5;                  // 0..1023 (32 rows each)
    const int colGrp  = job & 31;                  // 0..31   (64 cols each)
    const int r       = lane & 15;
    const int khalf   = lane >> 4;
    const int row0    = rowPair * 32;

    const float* __restrict__ xr0 = x   + (size_t)(row0 +      r) * EMB;
    const float* __restrict__ xr1 = x   + (size_t)(row0 + 16 + r) * EMB;
    const float* __restrict__ e0  = emb + (size_t)(colGrp * 64 +  0 + r) * EMB;
    const float* __restrict__ e1  = emb + (size_t)(colGrp * 64 + 16 + r) * EMB;
    const float* __restrict__ e2  = emb + (size_t)(colGrp * 64 + 32 + r) * EMB;
    const float* __restrict__ e3  = emb + (size_t)(colGrp * 64 + 48 + r) * EMB;

    v8f c00 = {}, c01 = {}, c02 = {}, c03 = {};
    v8f c10 = {}, c11 = {}, c12 = {}, c13 = {};
    const int koff = 2 * khalf;
    for (int k = 0; k < EMB; k += 4) {
        const int kk = k + koff;
        v2f a0 = *(const v2f*)(xr0 + kk);
        v2f a1 = *(const v2f*)(xr1 + kk);
        v2f b0 = *(const v2f*)(e0 + kk);
        v2f b1 = *(const v2f*)(e1 + kk);
        v2f b2 = *(const v2f*)(e2 + kk);
        v2f b3 = *(const v2f*)(e3 + kk);
        c00 = __builtin_amdgcn_wmma_f32_16x16x4_f32(false, a0, false, b0, (short)0, c00, false, false);
        c01 = __builtin_amdgcn_wmma_f32_16x16x4_f32(false, a0, false, b1, (short)0, c01, false, false);
        c02 = __builtin_amdgcn_wmma_f32_16x16x4_f32(false, a0, false, b2, (short)0, c02, false, false);
        c03 = __builtin_amdgcn_wmma_f32_16x16x4_f32(false, a0, false, b3, (short)0, c03, false, false);
        c10 = __builtin_amdgcn_wmma_f32_16x16x4_f32(false, a1, false, b0, (short)0, c10, false, false);
        c11 = __builtin_amdgcn_wmma_f32_16x16x4_f32(false, a1, false, b1, (short)0, c11, false, false);
        c12 = __builtin_amdgcn_wmma_f32_16x16x4_f32(false, a1, false, b2, (short)0, c12, false, false);
        c13 = __builtin_amdgcn_wmma_f32_16x16x4_f32(false, a1, false, b3, (short)0, c13, false, false);
    }

    float xs0[8], xs1[8];
    #pragma unroll
    for (int v = 0; v < 8; ++v) {
        xs0[v] = xsq[row0 +      v + 8 * khalf];
        xs1[v] = xsq[row0 + 16 + v + 8 * khalf];
    }

    #pragma unroll
    for (int tI = 0; tI < 4; ++tI) {
        v8f cva = (tI == 0) ? c00 : (tI == 1) ? c01 : (tI == 2) ? c02 : c03;
        v8f cvb = (tI == 0) ? c10 : (tI == 1) ? c11 : (tI == 2) ? c12 : c13;
        const int col = colGrp * 64 + tI * 16 + r;
        const float eq = esq[col];
        #pragma unroll
        for (int v = 0; v < 8; ++v) {
            const int orow0 = row0 +      v + 8 * khalf;
            const int orow1 = row0 + 16 + v + 8 * khalf;
            const float d0 = xs0[v] + eq - 2.0f * cva[v];
            const float d1 = xs1[v] + eq - 2.0f * cvb[v];
            enc[(size_t)orow0 * NCLS + col] = 1.0f / d0;
            enc[(size_t)orow1 * NCLS + col] = 1.0f / d1;
        }
    }
}

// ---------------------------------------------------------------------------
// Kernel 3: per row — group argmin (on sum of d), top-10 of 1/d in the group,
// L1-normalized weights, histogram of top-1. One 256-thread block per row.
// ---------------------------------------------------------------------------
__global__ __launch_bounds__(256)
void select_kernel(const float* __restrict__ enc, int* __restrict__ idxOut,
                   float* __restrict__ wOut, unsigned int* __restrict__ hist)
{
    const int row = blockIdx.x;
    const int t   = threadIdx.x;
    const float* __restrict__ er = enc + (size_t)row * NCLS;

    __shared__ float sv[256];
    __shared__ int   si[256];
    __shared__ float gsum[NGRP];
    __shared__ int   gsel;
    __shared__ float topv[NSAMP];
    __shared__ int   topi[NSAMP];

    // class g*256 + t  ->  group g; coalesced loads
    float dv[NGRP];
    #pragma unroll
    for (int g = 0; g < NGRP; ++g) dv[g] = 1.0f / er[g * CPG + t];

    // deterministic tree reduction per group
    for (int g = 0; g < NGRP; ++g) {
        sv[t] = dv[g];
        __syncthreads();
        for (int s = 128; s > 0; s >>= 1) { if (t < s) sv[t] += sv[t + s]; __syncthreads(); }
        if (t == 0) gsum[g] = sv[0];
        __syncthreads();
    }
    if (t == 0) {
        int best = 0; float bv = gsum[0];
        for (int g = 1; g < NGRP; ++g) if (gsum[g] < bv) { bv = gsum[g]; best = g; }
        gsel = best;
    }
    __syncthreads();
    const int g = gsel;
    float p = er[g * CPG + t];     // prob candidate held by this thread

    for (int it = 0; it < NSAMP; ++it) {
        sv[t] = p; si[t] = t;
        __syncthreads();
        for (int s = 128; s > 0; s >>= 1) {
            if (t < s && sv[t + s] > sv[t]) { sv[t] = sv[t + s]; si[t] = si[t + s]; }
            __syncthreads();
        }
        if (t == 0) { topv[it] = sv[0]; topi[it] = si[0]; }
        if (t == si[0]) p = -3.0e38f;   // knock out the winner
        __syncthreads();
    }

    if (t == 0) {
        float wsum = 0.0f;
        for (int it = 0; it < NSAMP; ++it) wsum += topv[it];
        const float norm = fmaxf(wsum, 1e-12f);
        for (int it = 0; it < NSAMP; ++it) {
            idxOut[row * NSAMP + it] = g * CPG + topi[it];
            wOut[row * NSAMP + it]   = topv[it] / norm;
        }
        atomicAdd(&hist[g * CPG + topi[0]], 1u);
    }
}

// ---------------------------------------------------------------------------
// Kernel 4: out0 = sum_s w[s]*emb[idx[s]]; accumulate SSE vs x (f64 atomic)
// ---------------------------------------------------------------------------
__global__ __launch_bounds__(128)
void gather_kernel(const float* __restrict__ x, const float* __restrict__ emb,
                   const int* __restrict__ idx, const float* __restrict__ w,
                   float* __restrict__ out0, double* __restrict__ sse)
{
    const int row = blockIdx.x;
    const int t   = threadIdx.x;
    __shared__ int   sI[NSAMP];
    __shared__ float sW[NSAMP];
    if (t < NSAMP) { sI[t] = idx[row * NSAMP + t]; sW[t] = w[row * NSAMP + t]; }
    __syncthreads();

    float local = 0.0f;
    for (int c = t; c < EMB; c += 128) {
        float acc = 0.0f;
        #pragma unroll
        for (int s = 0; s < NSAMP; ++s) acc += sW[s] * emb[(size_t)sI[s] * EMB + c];
        out0[(size_t)row * EMB + c] = acc;
        const float diff = x[(size_t)row * EMB + c] - acc;
        local += diff * diff;
    }
    __shared__ float red[128];
    red[t] = local; __syncthreads();
    for (int s = 64; s > 0; s >>= 1) { if (t < s) red[t] += red[t + s]; __syncthreads(); }
    if (t == 0) atomicAdd(sse, (double)red[0]);
}

// ---------------------------------------------------------------------------
// Kernel 5: entropy of top-1 histogram + loss = 1.25 * mse
// ---------------------------------------------------------------------------
__global__ __launch_bounds__(256)
void finalize_kernel(const unsigned int* __restrict__ hist, const double* __restrict__ sse,
                     float* __restrict__ lossOut, float* __restrict__ entOut)
{
    const int t = threadIdx.x;
    float local = 0.0f;
    for (int j = t; j < NCLS; j += 256) {
        const unsigned int h = hist[j];
        if (h > 0u) {
            const float pr = (float)h / (float)M_ROWS;
            local -= pr * logf(pr);
        }
    }
    __shared__ float red[256];
    red[t] = local; __syncthreads();
    for (int s = 128; s > 0; s >>= 1) { if (t < s) red[t] += red[t + s]; __syncthreads(); }
    if (t == 0) {
        *entOut  = red[0];
        *lossOut = 1.25f * (float)(*sse / (double)((size_t)M_ROWS * EMB));
    }
}

// ---------------------------------------------------------------------------
extern "C" void kernel_launch(void* const* d_in, const int* in_sizes, int n_in,
                              void* d_out, int out_size, void* d_ws, size_t ws_size,
                              hipStream_t stream)
{
    (void)in_sizes; (void)n_in; (void)out_size; (void)ws_size;
    const float* x   = (const float*)d_in[0];   // [32768, 400]
    const float* emb = (const float*)d_in[1];   // [2048, 400]
    float* out = (float*)d_out;

    // output layout: [loss][out0 13107200][entropy][encodings 67108864]
    float* lossPtr = out;
    float* out0    = out + 1;
    float* entPtr  = out + 1 + (size_t)M_ROWS * EMB;
    float* enc     = out + 2 + (size_t)M_ROWS * EMB;

    // workspace layout (~2.77 MB)
    char* ws = (char*)d_ws;
    float*        xsq  = (float*)(ws);                    // 32768 f
    float*        esq  = (float*)(ws + 131072);           // 2048 f
    int*          idx  = (int*)  (ws + 139264);           // 32768*10 i32
    float*        wgt  = (float*)(ws + 1449984);          // 32768*10 f
    unsigned int* hist = (unsigned int*)(ws + 2760704);   // 2048 u32
    double*       sse  = (double*)(ws + 2768896);         // 1 f64 (8B aligned)

    init_kernel<<<8, 256, 0, stream>>>(hist, sse);
    rowsq_kernel<<<M_ROWS, 128, 0, stream>>>(x,   xsq, EMB);
    rowsq_kernel<<<NCLS,   128, 0, stream>>>(emb, esq, EMB);
    // 1024 row-pairs (32 rows) * 32 col-groups (64 cols) = 32768 wave-jobs / 8
    gemm_dist_kernel<<<4096, 256, 0, stream>>>(x, emb, xsq, esq, enc);
    select_kernel<<<M_ROWS, 256, 0, stream>>>(enc, idx, wgt, hist);
    gather_kernel<<<M_ROWS, 128, 0, stream>>>(x, emb, idx, wgt, out0, sse);
    finalize_kernel<<<1, 256, 0, stream>>>(hist, sse, lossPtr, entPtr);
}